// GaussianTransformLayerC_86955907874862
// MI455X (gfx1250) — compile-verified
//
#include <hip/hip_runtime.h>

typedef float v2f __attribute__((ext_vector_type(2)));
typedef float v4f __attribute__((ext_vector_type(4)));
typedef float v8f __attribute__((ext_vector_type(8)));

#define CWT_S 8192
#define CWT_C 64
#define CWT_B 32
#define PI_D 3.14159265358979323846

struct F3 { float a, b, c; };   // 12B packed triple -> encourages b96 store

// ---------------------------------------------------------------------------
// Init kernel: reproduce pywt gaus1 filter construction (precision 10) and
// bake banded-Toeplitz A-matrix fragments in WMMA 16x4-f32 lane layout.
// Afrag layout in ws (float2): [scale(3)][chunk(20)][lane(32)] -> (A[m][kb], A[m][kb+1])
//   m = lane & 15, kb = 4*chunk + 2*(lane>>4)
//   A[m][r] = w[r - 31 + ofs - m] if 0 <= j < T else 0
// ---------------------------------------------------------------------------
__global__ void cwt_init_taps(float2* __restrict__ afrag) {
    __shared__ double s_intpsi[1024];
    __shared__ float  s_w[3][64];

    if (threadIdx.x == 0) {
        const double step = 10.0 / 1023.0;
        const double c4 = pow(2.0 / PI_D, 0.25);
        double acc = 0.0;
        for (int i = 0; i < 1024; ++i) {
            double xv = -5.0 + (10.0 * (double)i) / 1023.0;
            double psi = -2.0 * xv * exp(-xv * xv) * c4;
            acc += psi;
            s_intpsi[i] = acc * step;
        }
        const int Ls[3] = {11, 31, 61};
        const int as[3] = {1, 3, 6};
        for (int si = 0; si < 3; ++si) {
            const int L = Ls[si];
            const int a = as[si];
            float k[61];
            for (int t = 0; t < L; ++t) {
                int j = (int)((double)t / ((double)a * step));
                if (j > 1023) j = 1023;
                k[t] = (float)s_intpsi[j];
            }
            const float sa = sqrtf((float)a);
            for (int j = 0; j <= L; ++j) {
                float km1 = (j > 0) ? k[j - 1] : 0.0f;
                float kj  = (j < L) ? k[j]     : 0.0f;
                s_w[si][j] = -sa * (km1 - kj);   // taps for cross-correlation
            }
            for (int j = L + 1; j < 64; ++j) s_w[si][j] = 0.0f;
        }
    }
    __syncthreads();

    const int ofs[3] = {6, 16, 31};
    const int Ts[3]  = {12, 32, 62};
    for (int e = threadIdx.x; e < 3 * 20 * 32; e += blockDim.x) {
        const int lane = e & 31;
        const int q    = (e >> 5) % 20;
        const int si   = e / (20 * 32);
        const int m    = lane & 15;
        const int kb   = 4 * q + 2 * (lane >> 4);
        const int j0   = kb     - 31 + ofs[si] - m;
        const int j1   = kb + 1 - 31 + ofs[si] - m;
        float2 v;
        v.x = (j0 >= 0 && j0 < Ts[si]) ? s_w[si][j0] : 0.0f;
        v.y = (j1 >= 0 && j1 < Ts[si]) ? s_w[si][j1] : 0.0f;
        afrag[e] = v;
    }
}

// ---------------------------------------------------------------------------
// Main kernel: block = 256 threads (8 waves). Block covers 128 s-positions x
// 16 channels for one batch. LDS slab: rows r=0..191 -> x[s0-31+r][c0..c0+15],
// filled with b128 loads/stores. Each wave computes one 16(s)x16(c) tile for
// all 3 scales with exact-f32 WMMA (V_WMMA_F32_16X16X4_F32).
// ---------------------------------------------------------------------------
__global__ __launch_bounds__(256) void cwt_wmma_kernel(
    const float* __restrict__ x,
    float* __restrict__ out,
    const float2* __restrict__ afrag) {

    __shared__ float lds[192 * 16];

    const int s0 = blockIdx.x * 128;   // 64 blocks along S
    const int c0 = blockIdx.y * 16;    // 4 channel tiles
    const int b  = blockIdx.z;         // 32 batches
    const int tid = threadIdx.x;

    // -------- stage input slab, vectorized (zero-padded at S boundaries) ----
    const float* xb = x + (size_t)b * CWT_S * CWT_C + c0;
    #pragma unroll
    for (int it = 0; it < 3; ++it) {
        const int e4  = tid + it * 256;      // 0..767 quad-index
        const int row = e4 >> 2;             // 0..191
        const int c4  = (e4 & 3) * 4;        // 0,4,8,12
        const int s   = s0 - 31 + row;
        v4f v = {};
        if (s >= 0 && s < CWT_S)
            v = *(const v4f*)(xb + (size_t)s * CWT_C + c4);
        *(v4f*)(lds + row * 16 + c4) = v;
    }
    __syncthreads();

    const int wave = tid >> 5;
    const int lane = tid & 31;
    const int hi   = lane >> 4;   // 0 or 1
    const int n    = lane & 15;
    const int base_row = wave * 16;

    v8f acc0 = {}, acc1 = {}, acc2 = {};

    const float2* af0 = afrag;                 // scale 1: chunks 6..12
    const float2* af1 = afrag + 1 * 20 * 32;   // scale 3: chunks 3..15
    const float2* af2 = afrag + 2 * 20 * 32;   // scale 6: chunks 0..19

    for (int q = 0; q < 20; ++q) {
        const int rb  = 4 * q + 2 * hi;
        const int off = (base_row + rb) * 16 + n;
        v2f bfrag;
        bfrag.x = lds[off];
        bfrag.y = lds[off + 16];

        if (q >= 6 && q <= 12) {                     // wave-uniform branch
            float2 a = af0[q * 32 + lane];
            v2f av = {a.x, a.y};
            acc0 = __builtin_amdgcn_wmma_f32_16x16x4_f32(
                false, av, false, bfrag, (short)0, acc0, false, false);
        }
        if (q >= 3 && q <= 15) {                     // wave-uniform branch
            float2 a = af1[q * 32 + lane];
            v2f av = {a.x, a.y};
            acc1 = __builtin_amdgcn_wmma_f32_16x16x4_f32(
                false, av, false, bfrag, (short)0, acc1, false, false);
        }
        {
            float2 a = af2[q * 32 + lane];
            v2f av = {a.x, a.y};
            acc2 = __builtin_amdgcn_wmma_f32_16x16x4_f32(
                false, av, false, bfrag, (short)0, acc2, false, false);
        }
    }

    // -------- store: out[b][s][c*3 + scale], 3 scales packed per (s,c) -----
    const int s0w = s0 + wave * 16;
    const size_t outbase = (size_t)b * CWT_S * (3 * CWT_C);
    #pragma unroll
    for (int v = 0; v < 8; ++v) {
        const int m = v + 8 * hi;                 // C/D layout: M = vgpr + 8*(lane>=16)
        const size_t o = outbase + (size_t)(s0w + m) * (3 * CWT_C) + (size_t)(c0 + n) * 3;
        F3 t; t.a = acc0[v]; t.b = acc1[v]; t.c = acc2[v];
        *(F3*)(out + o) = t;
    }
}

extern "C" void kernel_launch(void* const* d_in, const int* in_sizes, int n_in,
                              void* d_out, int out_size, void* d_ws, size_t ws_size,
                              hipStream_t stream) {
    (void)in_sizes; (void)n_in; (void)out_size; (void)ws_size;
    const float* x = (const float*)d_in[0];
    float* out = (float*)d_out;
    float2* afrag = (float2*)d_ws;   // 1920 float2 = 15360 bytes

    cwt_init_taps<<<dim3(1), dim3(256), 0, stream>>>(afrag);

    dim3 grid(CWT_S / 128, CWT_C / 16, CWT_B);
    cwt_wmma_kernel<<<grid, dim3(256), 0, stream>>>(x, out, afrag);
}